// CRF_Gaussian_48095043781146
// MI455X (gfx1250) — compile-verified
//
#include <hip/hip_runtime.h>
#include <hip/hip_bf16.h>
#include <stdint.h>

#define D_DIM        512
#define CHUNK        8
#define SPMM_THREADS 128
#define NUM_ITERS    10

typedef __attribute__((ext_vector_type(2))) float v2f;
typedef __attribute__((ext_vector_type(8))) float v8f;

// ---------- CDNA5 async global->LDS helpers ----------
__device__ __forceinline__ uint32_t lds_offset(const void* p) {
  typedef __attribute__((address_space(3))) const char* lds_cptr;
  return (uint32_t)(unsigned long long)(lds_cptr)p;
}

__device__ __forceinline__ void async_load_b128(uint32_t lds_addr, const void* gaddr) {
  // per-lane: copies 16 bytes from the global address in gaddr to the LDS address in lds_addr, tracked by ASYNCcnt
  asm volatile("global_load_async_to_lds_b128 %0, %1, off"
               :: "v"(lds_addr), "v"(gaddr) : "memory");
}
__device__ __forceinline__ void wait_async_le8() {
  asm volatile("s_wait_asynccnt 0x8" ::: "memory");
}
__device__ __forceinline__ void wait_async_0() {
  asm volatile("s_wait_asynccnt 0x0" ::: "memory");
}

__device__ __forceinline__ int imin(int a, int b) { return a < b ? a : b; }

__device__ __forceinline__ float v8_get(v8f v, int i) {
  float r = v[0];
  r = (i == 1) ? v[1] : r;
  r = (i == 2) ? v[2] : r;
  r = (i == 3) ? v[3] : r;
  r = (i == 4) ? v[4] : r;
  r = (i == 5) ? v[5] : r;
  r = (i == 6) ? v[6] : r;
  r = (i == 7) ? v[7] : r;
  return r;
}

// ---------- CSR row_ptr from sorted edge_row (binary search per row) ----------
__global__ void rowptr_kernel(const int* __restrict__ erow, int* __restrict__ rp,
                              int n, int nnz) {
  int r = blockIdx.x * blockDim.x + threadIdx.x;
  if (r > n) return;
  if (r == n) { rp[n] = nnz; return; }
  int lo = 0, hi = nnz;
  while (lo < hi) { int mid = (lo + hi) >> 1; if (erow[mid] < r) lo = mid + 1; else hi = mid; }
  rp[r] = lo;
}

// ---------- per-row inverse L2 norm (wave per row) ----------
__global__ __launch_bounds__(256) void norm_kernel(const float* __restrict__ x,
                                                   float* __restrict__ invn, int n) {
  const int wid = threadIdx.x >> 5, lane = threadIdx.x & 31;
  const int r = blockIdx.x * 8 + wid;
  if (r >= n) return;
  const float4* row = (const float4*)(x + (size_t)r * D_DIM);
  float s = 0.f;
#pragma unroll
  for (int j = 0; j < 4; ++j) {
    float4 v = row[lane + 32 * j];
    s += v.x * v.x + v.y * v.y + v.z * v.z + v.w * v.w;
  }
#pragma unroll
  for (int off = 16; off; off >>= 1) s += __shfl_xor(s, off, 32);
  if (lane == 0) invn[r] = rsqrtf(fmaxf(s, 1e-12f));
}

// ---------- per-edge support values via WMMA (16 edges per wave) ----------
// A (16x4 f32): lane L holds M=L%16, VGPR j holds K=2*(L/16)+j -> load row[kb+2h..+1]
// B (4x16 f32): lane L holds N=L%16, same K striping          -> load col[kb+2h..+1]
// C (16x16 f32): diag(m) in VGPR m&7, lane m (m<8) / lane m+16 (m>=8)
__global__ __launch_bounds__(256) void sup_wmma_kernel(
    const float* __restrict__ x, const float* __restrict__ invn,
    const float* __restrict__ edge_val, const int* __restrict__ erow,
    const int* __restrict__ ecol, const float* __restrict__ sigma,
    float* __restrict__ sup, int nnz) {
  const int wid = threadIdx.x >> 5, lane = threadIdx.x & 31;
  const int base = (blockIdx.x * 8 + wid) * 16;   // 16 edges per wave
  if (base >= nnz) return;                        // uniform per wave -> EXEC stays full

  const int m = lane & 15;
  const int h = lane >> 4;
  int e = base + m;
  if (e >= nnz) e = nnz - 1;                      // clamp loads for tail group
  const int r = erow[e], c = ecol[e];
  const float* rptr = x + (size_t)r * D_DIM + 2 * h;
  const float* cptr = x + (size_t)c * D_DIM + 2 * h;

  v8f acc = {};
#pragma unroll 8
  for (int kb = 0; kb < D_DIM; kb += 4) {
    v2f av = *(const v2f*)(rptr + kb);
    v2f bv = *(const v2f*)(cptr + kb);
    acc = __builtin_amdgcn_wmma_f32_16x16x4_f32(
        /*neg_a=*/false, av, /*neg_b=*/false, bv,
        /*c_mod=*/(short)0, acc, /*reuse_a=*/false, /*reuse_b=*/false);
  }

  // extract diagonal: lanes 0..7 own m=lane (vgpr=lane); lanes 24..31 own m=lane-16 (vgpr=lane-24)
  const bool writer = (lane < 8) || (lane >= 24);
  const int  idx    = (lane < 8) ? lane : (lane - 24);
  const float d     = v8_get(acc, idx);
  if (writer && (base + m) < nnz) {
    float sim = (r == c) ? 0.f : d * invn[r] * invn[c];
    float s = expf(sigma[0]);
    // reference: exp(sim / (2 * (2*s*s))) = exp(sim * 0.25 / s^2)
    sup[base + m] = edge_val[base + m] * expf(sim * (0.25f / (s * s)));
  }
}

// ---------- per-row 1/denom ----------
__global__ void denom_kernel(const float* __restrict__ sup, const int* __restrict__ rp,
                             const float* __restrict__ alpha, const float* __restrict__ beta,
                             float* __restrict__ inv_denom, int n) {
  int r = blockIdx.x * blockDim.x + threadIdx.x;
  if (r >= n) return;
  float s = 0.f;
  const int e0 = rp[r], e1 = rp[r + 1];
  for (int e = e0; e < e1; ++e) s += sup[e];
  const float a = expf(alpha[0]);
  const float b = expf(beta[0]);
  inv_denom[r] = 1.0f / (b + s * a + a);
}

// ---------- fused SpMM + update, async-gather double-buffered through LDS ----------
__global__ __launch_bounds__(SPMM_THREADS) void spmm_kernel(
    const float* __restrict__ src, const float* __restrict__ x,
    const float* __restrict__ sup, const int* __restrict__ rp,
    const int* __restrict__ ecol, const float* __restrict__ inv_denom,
    const float* __restrict__ alpha, const float* __restrict__ beta,
    float* __restrict__ dst) {
  // 2 buffers * 8 rows * 2KB + 2KB dump area = 34KB LDS
  __shared__ __align__(16) float4 smem[2 * CHUNK * SPMM_THREADS + SPMM_THREADS];

  const int r = blockIdx.x;
  const int t = threadIdx.x;                 // owns floats [4t, 4t+4)
  const int start = rp[r], end = rp[r + 1];
  const int deg = end - start;

  const uint32_t base_off = lds_offset(smem);
  const uint32_t my16     = (uint32_t)t * 16u;
  const uint32_t dump_off = base_off + (uint32_t)(2 * CHUNK * SPMM_THREADS) * 16u + my16;

  float4 acc = make_float4(0.f, 0.f, 0.f, 0.f);
  const int nch = (deg + CHUNK - 1) / CHUNK;

  auto issue = [&](int ch, int buf) {
    const int base = start + ch * CHUNK;
    const int cnt  = imin(CHUNK, end - base);
#pragma unroll
    for (int j = 0; j < CHUNK; ++j) {
      int c;
      uint32_t lds;
      if (j < cnt) {
        c   = ecol[base + j];
        lds = base_off + (uint32_t)((buf * CHUNK + j) * SPMM_THREADS) * 16u + my16;
      } else {                       // pad so every chunk issues exactly 8 async ops
        c   = r;
        lds = dump_off;
      }
      const char* g = (const char*)src + ((size_t)c << 11) + my16; // row stride 2KB
      async_load_b128(lds, g);
    }
  };

  int buf = 0;
  if (nch > 0) issue(0, 0);
  for (int ch = 0; ch < nch; ++ch) {
    if (ch + 1 < nch) { issue(ch + 1, buf ^ 1); wait_async_le8(); }
    else              { wait_async_0(); }
    const int base = start + ch * CHUNK;
    const int cnt  = imin(CHUNK, end - base);
    for (int j = 0; j < cnt; ++j) {
      const float w = sup[base + j];
      float4 v = smem[(buf * CHUNK + j) * SPMM_THREADS + t]; // slice this thread wrote
      acc.x = fmaf(w, v.x, acc.x);
      acc.y = fmaf(w, v.y, acc.y);
      acc.z = fmaf(w, v.z, acc.z);
      acc.w = fmaf(w, v.w, acc.w);
    }
    buf ^= 1;
  }

  const float a   = expf(alpha[0]);
  const float b   = expf(beta[0]);
  const float idn = inv_denom[r];
  const size_t rowoff = ((size_t)r << 9) + (size_t)(t << 2);
  const float4 xv = *(const float4*)(x + rowoff);
  const float4 pv = *(const float4*)(src + rowoff);
  float4 o;
  o.x = (xv.x * b + (acc.x + pv.x) * a) * idn;
  o.y = (xv.y * b + (acc.y + pv.y) * a) * idn;
  o.z = (xv.z * b + (acc.z + pv.z) * a) * idn;
  o.w = (xv.w * b + (acc.w + pv.w) * a) * idn;
  *(float4*)(dst + rowoff) = o;
}

// ---------- host launcher ----------
extern "C" void kernel_launch(void* const* d_in, const int* in_sizes, int n_in,
                              void* d_out, int out_size, void* d_ws, size_t ws_size,
                              hipStream_t stream) {
  const float* x        = (const float*)d_in[0];
  const float* alpha    = (const float*)d_in[1];
  const float* beta     = (const float*)d_in[2];
  const float* sigma    = (const float*)d_in[3];
  const float* edge_val = (const float*)d_in[4];
  const int*   edge_row = (const int*)d_in[5];
  const int*   edge_col = (const int*)d_in[6];
  float* out = (float*)d_out;

  const int N   = in_sizes[0] / D_DIM;
  const int NNZ = in_sizes[4];

  // workspace carve (256B aligned)
  char* w = (char*)d_ws;
  auto carve = [&](size_t bytes) { char* p = w; w += (bytes + 255) & ~(size_t)255; return p; };
  int*   rp        = (int*)carve((size_t)(N + 1) * sizeof(int));
  float* invn      = (float*)carve((size_t)N * sizeof(float));
  float* sup       = (float*)carve((size_t)NNZ * sizeof(float));
  float* inv_denom = (float*)carve((size_t)N * sizeof(float));
  float* buf       = (float*)carve((size_t)N * D_DIM * sizeof(float));

  // one-time precompute
  rowptr_kernel<<<(N + 1 + 255) / 256, 256, 0, stream>>>(edge_row, rp, N, NNZ);
  norm_kernel<<<(N + 7) / 8, 256, 0, stream>>>(x, invn, N);
  const int ngroups = (NNZ + 15) / 16;      // 16 edges per wave, 8 waves per block
  sup_wmma_kernel<<<(ngroups + 7) / 8, 256, 0, stream>>>(x, invn, edge_val, edge_row,
                                                         edge_col, sigma, sup, NNZ);
  denom_kernel<<<(N + 255) / 256, 256, 0, stream>>>(sup, rp, alpha, beta, inv_denom, N);

  // 10 Jacobi iterations, ping-pong so iteration 10 writes d_out
  for (int k = 1; k <= NUM_ITERS; ++k) {
    float* dst = (k & 1) ? buf : out;
    const float* src = (k == 1) ? x : ((k & 1) ? (const float*)out : (const float*)buf);
    spmm_kernel<<<N, SPMM_THREADS, 0, stream>>>(src, x, sup, rp, edge_col, inv_denom,
                                                alpha, beta, dst);
  }
}